// GCN_83537113907565
// MI455X (gfx1250) — compile-verified
//
#include <hip/hip_runtime.h>
#include <math.h>

#define DD 128
#define NCONV 4
#define S2S_STEPS 3

typedef __attribute__((ext_vector_type(2))) float v2f;
typedef __attribute__((ext_vector_type(8))) float v8f;
typedef int b128_t __attribute__((vector_size(16)));   // matches builtin pointee type

// ---------- CDNA5 async global->LDS staging (ASYNCcnt path) ----------
#if defined(__gfx1250__) && __has_builtin(__builtin_amdgcn_global_load_async_to_lds_b128)
#define ASYNC_LDS 1
#else
#define ASYNC_LDS 0
#endif

// Stage nf4 float4's from global g into LDS l, 256-thread cooperative.
__device__ __forceinline__ void lds_stage(const float* __restrict__ g,
                                          float* __restrict__ l,
                                          int nf4, int tid) {
#if ASYNC_LDS
    for (int i = tid; i < nf4; i += 256) {
        __builtin_amdgcn_global_load_async_to_lds_b128(
            (__attribute__((address_space(1))) b128_t*)(g + (size_t)i * 4),
            (__attribute__((address_space(3))) b128_t*)(l + (size_t)i * 4),
            0, 0);
    }
#else
    for (int i = tid; i < nf4; i += 256)
        ((float4*)l)[i] = ((const float4*)g)[i];
#endif
}

__device__ __forceinline__ void async_join() {
#if ASYNC_LDS
#if __has_builtin(__builtin_amdgcn_s_wait_asynccnt)
    __builtin_amdgcn_s_wait_asynccnt(0);
#else
    asm volatile("s_wait_asynccnt 0" ::: "memory");
#endif
#endif
}

// ---------------- utility kernels ----------------
__global__ void zero_f32(float* __restrict__ p, int n) {
    int i = blockIdx.x * blockDim.x + threadIdx.x;
    if (i < n) p[i] = 0.f;
}

__global__ void fill_u32(unsigned* __restrict__ p, unsigned v, int n) {
    int i = blockIdx.x * blockDim.x + threadIdx.x;
    if (i < n) p[i] = v;
}

// ZeroPad2d: [N, FIN] -> [N, 128]
__global__ __launch_bounds__(DD) void pad_kernel(const float* __restrict__ x,
                                                 float* __restrict__ h,
                                                 int n, int fin) {
    int i = blockIdx.x;
    int d = threadIdx.x;
    h[(size_t)i * DD + d] = (d < fin) ? x[(size_t)i * fin + d] : 0.f;
}

__global__ void copy_f4(const float4* __restrict__ src, float4* __restrict__ dst, int n4) {
    int i = blockIdx.x * blockDim.x + threadIdx.x;
    if (i < n4) dst[i] = src[i];
}

// ---------------- GIN scatter-add: xacc[dst] += h[src], one wave per edge ----------------
__global__ __launch_bounds__(256) void scatter_kernel(const float* __restrict__ h,
                                                      float* __restrict__ xacc,
                                                      const int* __restrict__ srcs,
                                                      const int* __restrict__ dsts,
                                                      int ne) {
    int edge = (blockIdx.x * blockDim.x + threadIdx.x) >> 5;
    int lane = threadIdx.x & 31;
    if (edge >= ne) return;
    int s = srcs[edge];
    int d = dsts[edge];
    if (lane == 0 && edge + 8 < ne)            // hide gather latency of upcoming rows
        __builtin_prefetch(h + (size_t)srcs[edge + 8] * DD, 0, 1);
    const float4* hv = (const float4*)(h + (size_t)s * DD);
    float4 v = hv[lane];                       // 32 lanes x float4 = 128 floats
    float* out = xacc + (size_t)d * DD + lane * 4;
    atomicAdd(out + 0, v.x);
    atomicAdd(out + 1, v.y);
    atomicAdd(out + 2, v.z);
    atomicAdd(out + 3, v.w);
}

// ---------------- generic fp32 WMMA GEMM: C = [relu](A @ W^T + bias) ----------------
// A: [M, KT] row-major, W: [N, KT] row-major, C: [M, N].
// Grid: (ceil(M/16), N/128). Block: 256 threads = 8 waves; wave w owns 16-col tile.
// A strip (16 x KT) staged to LDS via async global->LDS; when STAGE_W, the 128-col
// weight panel (128 x KT) is staged too (K=128 -> 64 KB, 72 KB total of 320 KB/WGP).
template <int KT, bool STAGE_W>
__global__ __launch_bounds__(256) void wmma_gemm(const float* __restrict__ A,
                                                 const float* __restrict__ W,
                                                 const float* __restrict__ bias,
                                                 float* __restrict__ C,
                                                 int M, int N, int relu) {
    __shared__ alignas(16) float As[16 * KT];
    __shared__ alignas(16) float Ws[STAGE_W ? 128 * KT : 4];
    const int tid  = threadIdx.x;
    const int m0   = blockIdx.x * 16;
    const int nblk = blockIdx.y * 128;
    const bool full = (m0 + 16 <= M);

    if (full) {
        lds_stage(A + (size_t)m0 * KT, As, 16 * KT / 4, tid);
    } else {
        for (int i = tid; i < 16 * KT / 4; i += 256) {
            int rr = i / (KT / 4), cc = i % (KT / 4);
            float4 v = make_float4(0.f, 0.f, 0.f, 0.f);
            if (m0 + rr < M) v = ((const float4*)(A + (size_t)(m0 + rr) * KT))[cc];
            ((float4*)As)[i] = v;
        }
    }
    if (STAGE_W) lds_stage(W + (size_t)nblk * KT, Ws, 128 * KT / 4, tid);
    async_join();
    __syncthreads();

    const int lane   = tid & 31;
    const int waveId = tid >> 5;
    const int lcol   = waveId * 16 + (lane & 15);  // column within 128-col panel
    const int ncol   = nblk + lcol;                // global output column
    const int khalf  = (lane >> 4) << 1;           // K offset 0 / 2 per half-wave

    const float* arow = As + (size_t)(lane & 15) * KT;                 // A: M = lane&15
    const float* wrow = STAGE_W ? (Ws + (size_t)lcol * KT)             // B[k,n] = W[n,k]
                                : (W + (size_t)ncol * KT);

    v8f c = {};
#pragma unroll 4
    for (int kk = 0; kk < KT; kk += 4) {
        v2f a, b;
        a.x = arow[kk + khalf];
        a.y = arow[kk + khalf + 1];
        b.x = wrow[kk + khalf];
        b.y = wrow[kk + khalf + 1];
        // (neg_a, A, neg_b, B, c_mod, C, reuse_a, reuse_b)
        c = __builtin_amdgcn_wmma_f32_16x16x4_f32(false, a, false, b, (short)0, c,
                                                  false, false);
    }

    const int   rbase = m0 + ((lane >> 4) << 3);   // D: M = r + 8*(lane>=16)
    const float bj    = bias ? bias[ncol] : 0.f;
    if (full) {
#pragma unroll
        for (int r2 = 0; r2 < 8; ++r2) {
            float v = c[r2] + bj;
            if (relu) v = v > 0.f ? v : 0.f;
            C[(size_t)(rbase + r2) * N + ncol] = v;
        }
    } else {
#pragma unroll
        for (int r2 = 0; r2 < 8; ++r2) {
            if (rbase + r2 < M) {
                float v = c[r2] + bj;
                if (relu) v = v > 0.f ? v : 0.f;
                C[(size_t)(rbase + r2) * N + ncol] = v;
            }
        }
    }
}

// ---------------- Set2Set ----------------
__device__ __forceinline__ unsigned enc_f32(float f) {
    unsigned u = __float_as_uint(f);
    return (u & 0x80000000u) ? ~u : (u | 0x80000000u);
}
__device__ __forceinline__ float dec_f32(unsigned u) {
    return __uint_as_float((u & 0x80000000u) ? (u & 0x7fffffffu) : ~u);
}

// combine gate pre-activations (i,f,g,o per torch order) into (h, c)
__global__ __launch_bounds__(DD) void lstm_pointwise(const float* __restrict__ gA,
                                                     const float* __restrict__ gB,
                                                     float* __restrict__ hs,
                                                     float* __restrict__ cs) {
    const int g = blockIdx.x, d = threadIdx.x;
    const size_t base = (size_t)g * 4 * DD;
    float gi = gA[base + 0 * DD + d] + gB[base + 0 * DD + d];
    float gf = gA[base + 1 * DD + d] + gB[base + 1 * DD + d];
    float gg = gA[base + 2 * DD + d] + gB[base + 2 * DD + d];
    float go = gA[base + 3 * DD + d] + gB[base + 3 * DD + d];
    float ig = 1.f / (1.f + __expf(-gi));
    float fg = 1.f / (1.f + __expf(-gf));
    float og = 1.f / (1.f + __expf(-go));
    float cn = fg * cs[(size_t)g * DD + d] + ig * tanhf(gg);
    cs[(size_t)g * DD + d] = cn;
    hs[(size_t)g * DD + d] = og * tanhf(cn);
}

// e[v] = dot(h[v], q[batch[v]]); segment max via encoded-uint atomicMax
__global__ __launch_bounds__(256) void edot_kernel(const float* __restrict__ h,
                                                   const float* __restrict__ hs,
                                                   const int* __restrict__ batch,
                                                   float* __restrict__ e,
                                                   unsigned* __restrict__ menc,
                                                   int n) {
    int node = (blockIdx.x * blockDim.x + threadIdx.x) >> 5;
    int lane = threadIdx.x & 31;
    if (node >= n) return;
    int g = batch[node];
    float4 a = ((const float4*)(h  + (size_t)node * DD))[lane];
    float4 b = ((const float4*)(hs + (size_t)g    * DD))[lane];
    float p = a.x * b.x + a.y * b.y + a.z * b.z + a.w * b.w;
#pragma unroll
    for (int off = 16; off; off >>= 1) p += __shfl_xor(p, off, 32);
    if (lane == 0) {
        e[node] = p;
        atomicMax(menc + g, enc_f32(p));
    }
}

__global__ void expsum_kernel(float* __restrict__ e,
                              const unsigned* __restrict__ menc,
                              float* __restrict__ s,
                              const int* __restrict__ batch, int n) {
    int v = blockIdx.x * blockDim.x + threadIdx.x;
    if (v >= n) return;
    int g = batch[v];
    float ex = __expf(e[v] - dec_f32(menc[g]));
    e[v] = ex;
    atomicAdd(s + g, ex);
}

// r[g] += (ex/s[g]) * h[v]
__global__ __launch_bounds__(256) void wsum_kernel(const float* __restrict__ h,
                                                   const float* __restrict__ e,
                                                   const float* __restrict__ s,
                                                   const int* __restrict__ batch,
                                                   float* __restrict__ r, int n) {
    int node = (blockIdx.x * blockDim.x + threadIdx.x) >> 5;
    int lane = threadIdx.x & 31;
    if (node >= n) return;
    int g = batch[node];
    float a = e[node] / s[g];
    float4 v = ((const float4*)(h + (size_t)node * DD))[lane];
    float* out = r + (size_t)g * DD + lane * 4;
    atomicAdd(out + 0, a * v.x);
    atomicAdd(out + 1, a * v.y);
    atomicAdd(out + 2, a * v.z);
    atomicAdd(out + 3, a * v.w);
}

__global__ void qstar_kernel(const float* __restrict__ hs, const float* __restrict__ r,
                             float* __restrict__ qstar, int b) {
    int i = blockIdx.x * blockDim.x + threadIdx.x;
    if (i >= b * 2 * DD) return;
    int g = i / (2 * DD), j = i % (2 * DD);
    qstar[i] = (j < DD) ? hs[(size_t)g * DD + j] : r[(size_t)g * DD + (j - DD)];
}

// ---------------- host orchestration ----------------
extern "C" void kernel_launch(void* const* d_in, const int* in_sizes, int n_in,
                              void* d_out, int out_size, void* d_ws, size_t ws_size,
                              hipStream_t stream) {
    (void)n_in; (void)ws_size;
    const float* x      = (const float*)d_in[0];
    const int*   eidx   = (const int*)d_in[1];
    const int*   batch  = (const int*)d_in[2];
    const float* W_gin  = (const float*)d_in[4];
    const float* b_gin  = (const float*)d_in[5];
    const float* W_ih   = (const float*)d_in[6];
    const float* W_hh   = (const float*)d_in[7];
    const float* b_ih   = (const float*)d_in[8];
    const float* b_hh   = (const float*)d_in[9];
    const float* W1     = (const float*)d_in[10];
    const float* b1     = (const float*)d_in[11];
    const float* W2     = (const float*)d_in[12];
    const float* b2     = (const float*)d_in[13];
    float* out = (float*)d_out;

    const int Nn  = in_sizes[2];          // nodes (batch_index length)
    const int Ee  = in_sizes[1] / 2;      // edges
    const int FIN = in_sizes[0] / Nn;     // raw feature width
    const int Bb  = out_size / DD;        // graphs

    // workspace carve-out (256B aligned)
    char* ws = (char*)d_ws;
    size_t off = 0;
    auto carve = [&](size_t bytes) -> char* {
        char* p = ws + off;
        off += (bytes + 255) & ~((size_t)255);
        return p;
    };
    float*    hbuf   = (float*)carve((size_t)Nn * DD * 4);
    float*    xbuf   = (float*)carve((size_t)Nn * DD * 4);
    float*    hsb    = (float*)carve((size_t)Bb * DD * 4);
    float*    csb    = (float*)carve((size_t)Bb * DD * 4);
    float*    qstar  = (float*)carve((size_t)Bb * 2 * DD * 4);
    float*    gatesA = (float*)carve((size_t)Bb * 4 * DD * 4);
    float*    gatesB = (float*)carve((size_t)Bb * 4 * DD * 4);
    float*    rbuf   = (float*)carve((size_t)Bb * DD * 4);
    float*    ebuf   = (float*)carve((size_t)Nn * 4);
    unsigned* menc   = (unsigned*)carve((size_t)Bb * 4);
    float*    sbuf   = (float*)carve((size_t)Bb * 4);
    float*    tbuf   = (float*)carve((size_t)Bb * DD * 4);

    const int* srcs = eidx;        // edge_index[0]
    const int* dsts = eidx + Ee;   // edge_index[1]

    // 1) pad features to D
    pad_kernel<<<Nn, DD, 0, stream>>>(x, hbuf, Nn, FIN);

    // 2) 4x GIN conv with shared weights: x_acc = h + scatter(h), h = relu(x_acc W^T + b)
    const int n4     = Nn * DD / 4;
    const int mtiles = (Nn + 15) / 16;
    for (int it = 0; it < NCONV; ++it) {
        copy_f4<<<(n4 + 255) / 256, 256, 0, stream>>>((const float4*)hbuf, (float4*)xbuf, n4);
        scatter_kernel<<<(Ee + 7) / 8, 256, 0, stream>>>(hbuf, xbuf, srcs, dsts, Ee);
        wmma_gemm<128, true><<<dim3(mtiles, 1), 256, 0, stream>>>(
            xbuf, W_gin, b_gin, hbuf, Nn, DD, 1);
    }

    // 3) Set2Set readout
    const int btiles = (Bb + 15) / 16;
    zero_f32<<<(Bb * DD + 255) / 256, 256, 0, stream>>>(hsb, Bb * DD);
    zero_f32<<<(Bb * DD + 255) / 256, 256, 0, stream>>>(csb, Bb * DD);
    zero_f32<<<(Bb * 2 * DD + 255) / 256, 256, 0, stream>>>(qstar, Bb * 2 * DD);
    for (int st = 0; st < S2S_STEPS; ++st) {
        // gates = q_star@W_ih^T + b_ih + h@W_hh^T + b_hh  (two WMMA GEMMs + pointwise)
        wmma_gemm<256, false><<<dim3(btiles, 4 * DD / 128), 256, 0, stream>>>(
            qstar, W_ih, b_ih, gatesA, Bb, 4 * DD, 0);
        wmma_gemm<128, true><<<dim3(btiles, 4 * DD / 128), 256, 0, stream>>>(
            hsb, W_hh, b_hh, gatesB, Bb, 4 * DD, 0);
        lstm_pointwise<<<Bb, DD, 0, stream>>>(gatesA, gatesB, hsb, csb);

        fill_u32<<<(Bb + 255) / 256, 256, 0, stream>>>(menc, 0x007FFFFFu, Bb); // enc(-inf)
        zero_f32<<<(Bb + 255) / 256, 256, 0, stream>>>(sbuf, Bb);
        zero_f32<<<(Bb * DD + 255) / 256, 256, 0, stream>>>(rbuf, Bb * DD);
        edot_kernel<<<(Nn + 7) / 8, 256, 0, stream>>>(hbuf, hsb, batch, ebuf, menc, Nn);
        expsum_kernel<<<(Nn + 255) / 256, 256, 0, stream>>>(ebuf, menc, sbuf, batch, Nn);
        wsum_kernel<<<(Nn + 7) / 8, 256, 0, stream>>>(hbuf, ebuf, sbuf, batch, rbuf, Nn);
        qstar_kernel<<<(Bb * 2 * DD + 255) / 256, 256, 0, stream>>>(hsb, rbuf, qstar, Bb);
    }

    // 4) post-MP MLP via WMMA GEMMs
    wmma_gemm<256, false><<<dim3(btiles, 1), 256, 0, stream>>>(qstar, W1, b1, tbuf, Bb, DD, 1);
    wmma_gemm<128, true><<<dim3(btiles, 1), 256, 0, stream>>>(tbuf, W2, b2, out, Bb, DD, 0);
}